// NTupleNetwork_86191403696556
// MI455X (gfx1250) — compile-verified
//
#include <hip/hip_runtime.h>

// ---------------------------------------------------------------------------
// NTupleNetwork forward for MI455X (gfx1250), fp32 WMMA (16x16x4) throughout.
// One workgroup (4 waves / 128 threads, wave32) per batch element.
// x staged into LDS via CDNA5 async global->LDS loads (ASYNCcnt) when the
// toolchain exposes the builtin; falls back to b128 load + ds_store.
// ---------------------------------------------------------------------------

typedef __attribute__((ext_vector_type(2))) float v2f;
typedef __attribute__((ext_vector_type(8))) float v8f;
typedef int v4i __attribute__((vector_size(16)));   // matches builtin pointee type

#if defined(__HIP_DEVICE_COMPILE__) &&                                   \
    __has_builtin(__builtin_amdgcn_global_load_async_to_lds_b128) &&     \
    __has_builtin(__builtin_amdgcn_s_wait_asynccnt)
#define USE_ASYNC_LDS 1
#else
#define USE_ASYNC_LDS 0
#endif

__device__ __forceinline__ v8f wmma4(v2f a, v2f b, v8f c) {
    // D = A(16x4 f32) * B(4x16 f32) + C(16x16 f32)
    return __builtin_amdgcn_wmma_f32_16x16x4_f32(
        /*neg_a=*/false, a, /*neg_b=*/false, b,
        /*c_mod=*/(short)0, c, /*reuse_a=*/false, /*reuse_b=*/false);
}

constexpr int NR      = 32;                    // rows per batch (n)
constexpr int D       = 1024;                  // embedding dim
constexpr int HB_COLS = 36;                    // 33 padded to 36 (9 K-steps of 4)
constexpr int SM_H    = 33 * D;                // rows 0..31 = x, row 32 = inter
constexpr int SM_INV  = SM_H + NR * HB_COLS;   // inverse norms offset
constexpr int SM_FLOATS = SM_INV + NR;         // total dynamic LDS floats (~140KB)

__global__ void ntuple_fwd_kernel(const float* __restrict__ x,
                                  const float* __restrict__ W1,
                                  const float* __restrict__ b1,
                                  const float* __restrict__ W2,
                                  const float* __restrict__ b2,
                                  float* __restrict__ out)
{
    extern __shared__ float sm[];
    float* xbuf = sm;             // 33 x 1024 (row 32 = inter = sims flattened)
    float* hbuf = sm + SM_H;      // 32 x 36  (cols 33..35 zero-padded)
    float* invn = sm + SM_INV;    // 32

    const int b    = blockIdx.x;
    const int tid  = threadIdx.x;
    const int wave = tid >> 5;
    const int lane = tid & 31;
    const int half = lane >> 4;   // 0: lanes 0-15, 1: lanes 16-31
    const int l    = lane & 15;

    // ---- stage x[b] (32x1024 f32 = 128KB) into LDS ----
#if USE_ASYNC_LDS
    {
        const char* gp = (const char*)(x + (size_t)b * NR * D);
        for (int i = tid; i < NR * D / 4; i += blockDim.x) {
            __builtin_amdgcn_global_load_async_to_lds_b128(
                (__attribute__((address_space(1))) v4i*)(uintptr_t)(gp + (size_t)i * 16),
                (__attribute__((address_space(3))) v4i*)(unsigned)(uintptr_t)(xbuf + i * 4),
                /*offset=*/0, /*cpol=*/0);
        }
        __builtin_amdgcn_s_wait_asynccnt(0);
    }
#else
    {
        const float4* xg = (const float4*)(x + (size_t)b * NR * D);
        float4* xl = (float4*)xbuf;
        for (int i = tid; i < NR * D / 4; i += blockDim.x) xl[i] = xg[i];
    }
#endif
    __syncthreads();

    // ---- per-row inverse norms: inv[r] = 1 / max(||x_r||, 1e-8) ----
    for (int r = wave * 8; r < wave * 8 + 8; ++r) {
        const float* row = xbuf + r * D;
        float s = 0.f;
        for (int k = lane; k < D; k += 32) { float v = row[k]; s += v * v; }
        for (int off = 16; off; off >>= 1) s += __shfl_xor(s, off, 32);
        if (lane == 0) invn[r] = 1.0f / fmaxf(sqrtf(s), 1e-8f);
    }
    __syncthreads();

    // ---- GEMM1: S = x * x^T (K=1024), one 16x16 tile per wave; scale by
    //      inv[i]*inv[j] in the epilogue and store as "inter" row (row 32) ----
    {
        const int mt = wave >> 1, nt = wave & 1;
        const float* ra = xbuf + (mt * 16 + l) * D;  // A: row m, K along row
        const float* rb = xbuf + (nt * 16 + l) * D;  // B[k][n] = x[n][k] -> same shape
        v8f acc = {};
        #pragma unroll 8
        for (int k0 = 0; k0 < D; k0 += 4) {
            v2f a  = *(const v2f*)(ra + k0 + 2 * half);
            v2f bb = *(const v2f*)(rb + k0 + 2 * half);
            acc = wmma4(a, bb, acc);
        }
        float* inter = xbuf + 32 * D;
        const int   n    = nt * 16 + l;
        const float invb = invn[n];
        #pragma unroll
        for (int v = 0; v < 8; ++v) {
            const int m = mt * 16 + v + 8 * half;     // C layout: M = vgpr + 8*half
            inter[m * 32 + n] = acc[v] * invn[m] * invb;
        }
    }
    __syncthreads();

    // ---- GEMM2: h' = gelu(combined' * W1^T + b1), M=32 (rows 1..32), N=33->48, K=1024.
    //      Padded columns n>=33 read a clamped (valid, finite) W1 row; their
    //      results live only in those lanes' accumulators and are discarded. ----
    for (int t = wave; t < 6; t += 4) {
        const int mt = t % 2, nt = t / 2;
        const float* ra = xbuf + (mt * 16 + l + 1) * D;   // combined rows 1..32
        const int  n  = nt * 16 + l;
        const float* wb = W1 + (size_t)(n < 33 ? n : 0) * D;  // B[k][n] = W1[n][k]
        v8f acc = {};
        #pragma unroll 4
        for (int k0 = 0; k0 < D; k0 += 4) {
            v2f a  = *(const v2f*)(ra + k0 + 2 * half);
            v2f bb = *(const v2f*)(wb + k0 + 2 * half);
            acc = wmma4(a, bb, acc);
        }
        #pragma unroll
        for (int v = 0; v < 8; ++v) {
            const int m = mt * 16 + v + 8 * half;
            if (n < HB_COLS) {
                float hv = 0.f;
                if (n < 33) {
                    float z = acc[v] + b1[n];
                    hv = 0.5f * z * (1.0f + erff(z * 0.70710678118654752440f));
                }
                hbuf[m * HB_COLS + n] = hv;               // pad cols 33..35 with 0
            }
        }
    }
    __syncthreads();

    // ---- GEMM3: out = h' * W2^T + b2, M=32, N=1024, K=33 (padded to 36).
    //      k is the contraction dim -> keep the k<33 guard on B (OOB bits
    //      could be NaN and 0*NaN would propagate). ----
    float* outb = out + (size_t)b * NR * 1024;
    for (int t = wave; t < 128; t += 4) {
        const int mt = t & 1, nt = t >> 1;
        const int n  = nt * 16 + l;
        const float* ha = hbuf + (mt * 16 + l) * HB_COLS;
        const float* wr = W2 + (size_t)n * 33;            // B[k][n] = W2[n][k]
        v8f acc = {};
        #pragma unroll
        for (int k0 = 0; k0 < HB_COLS; k0 += 4) {
            const int k = k0 + 2 * half;
            v2f a = *(const v2f*)(ha + k);                // zero-padded, no guard
            v2f bb;
            bb.x = (k     < 33) ? wr[k]     : 0.f;
            bb.y = (k + 1 < 33) ? wr[k + 1] : 0.f;
            acc = wmma4(a, bb, acc);
        }
        const float bias = b2[n];
        #pragma unroll
        for (int v = 0; v < 8; ++v) {
            const int m = mt * 16 + v + 8 * half;
            outb[m * 1024 + n] = acc[v] + bias;
        }
    }
}

extern "C" void kernel_launch(void* const* d_in, const int* in_sizes, int n_in,
                              void* d_out, int out_size, void* d_ws, size_t ws_size,
                              hipStream_t stream) {
    const float* x  = (const float*)d_in[0];
    const float* W1 = (const float*)d_in[1];
    const float* b1 = (const float*)d_in[2];
    const float* W2 = (const float*)d_in[3];
    const float* b2 = (const float*)d_in[4];
    float* out = (float*)d_out;

    const int B = in_sizes[0] / (NR * D);     // 2048
    const size_t smem = (size_t)SM_FLOATS * sizeof(float);   // ~140 KB (<320KB/WGP)
    (void)hipFuncSetAttribute((const void*)ntuple_fwd_kernel,
                              hipFuncAttributeMaxDynamicSharedMemorySize, (int)smem);
    ntuple_fwd_kernel<<<dim3(B), dim3(128), smem, stream>>>(x, W1, b1, W2, b2, out);
}